// PointPillarsScatter_31121333027326
// MI455X (gfx1250) — compile-verified
//
#include <hip/hip_runtime.h>
#include <stdint.h>

// ---------------------------------------------------------------------------
// PointPillarsScatter for MI455X (gfx1250).
// Zero-FLOP, bandwidth-bound op. Strategy: invert the scatter into a gather so
// that all 256 MiB of output is written exactly once with coalesced 128B
// non-temporal stores, instead of 6.4M random write-allocated 4B scatters.
//   Pass 1: fill 1M-entry index map (d_ws) with -1            (4 MiB, NT b128)
//   Pass 2: idx[flat(b,y,x)] = max(p)  (atomic, deterministic) (tiny)
//   Pass 3: gather: thread = one (b,y,x) cell; loop c=0..63,
//           wave writes 32 consecutive x per channel plane -> coalesced.
//           Index tile staged via gfx1250 async global->LDS copy.
// ---------------------------------------------------------------------------

typedef float v4f __attribute__((ext_vector_type(4)));
typedef int   v4i __attribute__((ext_vector_type(4)));

#define CC   64
#define NBX  512
#define NBY  512
#define NB   4
#define NPIL 100000
#define PLANE ((size_t)NBX * (size_t)NBY)           // elements per (b,c) plane
#define CELLS ((size_t)NB * PLANE)                  // 1,048,576 cells

typedef __attribute__((address_space(1))) int gint_t;  // global
typedef __attribute__((address_space(3))) int lint_t;  // LDS

// ---- gfx1250 async global->LDS copy (ASYNCcnt path, CDNA5 ISA) ----------
__device__ __forceinline__ void async_load_i32_to_lds(const int* gsrc, int* lds_dst) {
#if __has_builtin(__builtin_amdgcn_global_load_async_to_lds_b32)
    __builtin_amdgcn_global_load_async_to_lds_b32((gint_t*)gsrc, (lint_t*)lds_dst,
                                                  /*offset=*/0, /*cpol=*/0);
#else
    uint32_t lds_addr = (uint32_t)(uintptr_t)(lint_t*)lds_dst;
    asm volatile("global_load_async_to_lds_b32 %0, %1, %2"
                 :
                 : "v"(lds_addr), "v"((uint32_t)0), "s"(gsrc)
                 : "memory");
#endif
}

__device__ __forceinline__ void wait_asynccnt0() {
#if __has_builtin(__builtin_amdgcn_s_wait_asynccnt)
    __builtin_amdgcn_s_wait_asynccnt(0);
#else
    asm volatile("s_wait_asynccnt 0" ::: "memory");
#endif
}

// ---- Pass 1: fill index map with -1 (128-bit NT stores) --------------------
__global__ void __launch_bounds__(256)
fill_idx_kernel(v4i* __restrict__ idx4, int n4) {
    int t = blockIdx.x * blockDim.x + threadIdx.x;
    if (t < n4) {
        v4i v = { -1, -1, -1, -1 };
        __builtin_nontemporal_store(v, idx4 + t);
    }
}

// ---- Pass 2: scatter pillar index (atomicMax -> deterministic winner) ------
__global__ void __launch_bounds__(256)
scatter_idx_kernel(const int* __restrict__ coords, int* __restrict__ idx, int P) {
    int p = blockIdx.x * blockDim.x + threadIdx.x;
    if (p < P) {
        int b = coords[4 * p + 0];
        int x = coords[4 * p + 1];
        int y = coords[4 * p + 2];
        int flat = (b * NBY + y) * NBX + x;
        atomicMax(&idx[flat], p);
    }
}

// ---- Pass 3: gather. One thread per (b,y,x) cell, loops over channels. -----
// Block of 256 covers 256 consecutive x (NX%256==0 -> same b,y per block).
// Per channel, a wave's 32 lanes hit 32 consecutive floats: coalesced 128B.
__global__ void __launch_bounds__(256)
gather_kernel(const float* __restrict__ pf, const int* __restrict__ idx,
              float* __restrict__ out) {
    __shared__ int smem[256];
    const int tid  = threadIdx.x;
    const int cell = blockIdx.x * 256 + tid;

    // Stage the 256-entry index tile through LDS via the async copy path.
    async_load_i32_to_lds(idx + cell, &smem[tid]);
    wait_asynccnt0();
    __syncthreads();

    const int p     = smem[tid];
    const int b     = cell >> 18;                 // cell / (NY*NX), NY*NX = 2^18
    const int rem   = cell & (int)(PLANE - 1);    // y*NX + x
    float* o        = out + (size_t)b * CC * PLANE + (size_t)rem;

    // Branch-free: empty cells read pillar row 0 (L0-resident) and mask to 0,
    // so waves never diverge across the 64-store streaming loop.
    const int   psafe = (p >= 0) ? p : 0;
    const float m     = (p >= 0) ? 1.0f : 0.0f;
    const v4f* row = (const v4f*)(pf + (size_t)psafe * CC);

#pragma unroll
    for (int cg = 0; cg < CC / 4; ++cg) {
        v4f v = row[cg] * m;
        float* oc = o + (size_t)(4 * cg) * PLANE;
        __builtin_nontemporal_store(v.x, oc);
        __builtin_nontemporal_store(v.y, oc + PLANE);
        __builtin_nontemporal_store(v.z, oc + 2 * PLANE);
        __builtin_nontemporal_store(v.w, oc + 3 * PLANE);
    }
}

// ---- Fallback (ws too small): zero output, then direct scatter -------------
__global__ void __launch_bounds__(256)
zero_out_kernel(v4f* __restrict__ out4, int n4) {
    int t = blockIdx.x * blockDim.x + threadIdx.x;
    if (t < n4) {
        v4f z = { 0.f, 0.f, 0.f, 0.f };
        __builtin_nontemporal_store(z, out4 + t);
    }
}

__global__ void __launch_bounds__(256)
direct_scatter_kernel(const float* __restrict__ pf, const int* __restrict__ coords,
                      float* __restrict__ out, int P) {
    int p = blockIdx.x * blockDim.x + threadIdx.x;
    if (p >= P) return;
    int b = coords[4 * p + 0];
    int x = coords[4 * p + 1];
    int y = coords[4 * p + 2];
    float* o = out + (size_t)b * CC * PLANE + (size_t)y * NBX + (size_t)x;
    const v4f* row = (const v4f*)(pf + (size_t)p * CC);
#pragma unroll
    for (int cg = 0; cg < CC / 4; ++cg) {
        v4f v = row[cg];
        float* oc = o + (size_t)(4 * cg) * PLANE;
        __builtin_nontemporal_store(v.x, oc);
        __builtin_nontemporal_store(v.y, oc + PLANE);
        __builtin_nontemporal_store(v.z, oc + 2 * PLANE);
        __builtin_nontemporal_store(v.w, oc + 3 * PLANE);
    }
}

// ---------------------------------------------------------------------------
extern "C" void kernel_launch(void* const* d_in, const int* in_sizes, int n_in,
                              void* d_out, int out_size, void* d_ws, size_t ws_size,
                              hipStream_t stream) {
    const float* pf     = (const float*)d_in[0];   // (P, 64) float32
    const int*   coords = (const int*)d_in[1];     // (P, 4)  int32 [b,x,y,z]
    float*       out    = (float*)d_out;           // (B, 64, NY, NX) float32
    const int P = in_sizes[1] / 4;                 // 100000

    const size_t idx_bytes = CELLS * sizeof(int);  // 4 MiB scratch needed

    if (ws_size >= idx_bytes) {
        int* idx = (int*)d_ws;
        int n4 = (int)(CELLS / 4);                                 // 262,144
        fill_idx_kernel<<<(n4 + 255) / 256, 256, 0, stream>>>((v4i*)idx, n4);
        scatter_idx_kernel<<<(P + 255) / 256, 256, 0, stream>>>(coords, idx, P);
        gather_kernel<<<(int)(CELLS / 256), 256, 0, stream>>>(pf, idx, out);
    } else {
        int n4 = out_size / 4;                                     // 16,777,216
        zero_out_kernel<<<(n4 + 255) / 256, 256, 0, stream>>>((v4f*)out, n4);
        direct_scatter_kernel<<<(P + 255) / 256, 256, 0, stream>>>(pf, coords, out, P);
    }
}